// GNNChoiceModel_23794118820072
// MI455X (gfx1250) — compile-verified
//
#include <hip/hip_runtime.h>
#include <hip/hip_bf16.h>

typedef __attribute__((ext_vector_type(2))) float v2f;
typedef __attribute__((ext_vector_type(8))) float v8f;
typedef int v4i __attribute__((vector_size(16)));   // matches builtin's pointee type
typedef __attribute__((address_space(1))) v4i* gv4i_p;  // global (HIP: __device__)
typedef __attribute__((address_space(3))) v4i* lv4i_p;  // LDS    (HIP: __shared__)

#define NEG_SLOPE 0.2f

// ---------- CDNA5 async global->LDS copy (ASYNCcnt-tracked) ----------
#if defined(__has_builtin) && __has_builtin(__builtin_amdgcn_global_load_async_to_lds_b128)
#define ASYNC_CP_B128(gp, lp)                                                  \
  __builtin_amdgcn_global_load_async_to_lds_b128(                              \
      (gv4i_p)(size_t)(gp), (lv4i_p)(unsigned)(size_t)(lp), 0, 0)
#else
#define ASYNC_CP_B128(gp, lp)                                                  \
  asm volatile("global_load_async_to_lds_b128 %0, %1, off" ::                  \
                   "v"((unsigned)(size_t)(lp)),                                \
               "v"((const void*)(gp))                                          \
               : "memory")
#endif

#if defined(__has_builtin) && __has_builtin(__builtin_amdgcn_s_wait_asynccnt)
#define WAIT_ASYNC0() __builtin_amdgcn_s_wait_asynccnt(0)
#else
#define WAIT_ASYNC0() asm volatile("s_wait_asynccnt 0" ::: "memory")
#endif

// ---------- helpers: monotone float<->uint for atomicMax on floats ----------
__device__ __forceinline__ unsigned fflip(float f) {
  unsigned u = __float_as_uint(f);
  unsigned mask = (unsigned)(-(int)(u >> 31)) | 0x80000000u;
  return u ^ mask;
}
__device__ __forceinline__ float funflip(unsigned u) {
  unsigned mask = ((u >> 31) - 1u) | 0x80000000u;
  return __uint_as_float(u ^ mask);
}
__device__ __forceinline__ float lrelu(float v) { return v > 0.f ? v : NEG_SLOPE * v; }

// ---------- GEMM: Hout[M,64] = X[M,64] @ W[64,64], fp32 WMMA 16x16x4 ----------
// Block = 256 threads = 8 waves -> one 32x64 output block (16x16 tile / wave).
// A-tile (32x64) and W (64x64) staged in LDS via GLOBAL_LOAD_ASYNC_TO_LDS_B128.
// A-tile LDS stride padded to 68 floats: lanes of a half-wave read rows r..r+15
// at the same k, bank = (r*68+k)%64 steps by 4 per row -> 16 distinct banks.
__global__ void __launch_bounds__(256) k_gemm64(const float* __restrict__ X,
                                                const float* __restrict__ W,
                                                float* __restrict__ Hout, int M) {
  __shared__ float Xs[32 * 68];  // 8.5 KB, padded (row stride 272B, 16B-aligned)
  __shared__ float Ws[64 * 64];  // 16 KB, lane-consecutive reads -> conflict-free

  const int tid  = threadIdx.x;
  const int row0 = blockIdx.x * 32;
  if (row0 + 32 > M) return;  // uniform per block: EXEC stays all-1s for WMMA

  // ---- async stage ----
  const float* gX = X + (size_t)row0 * 64;
  for (int c = tid; c < 512; c += 256) {  // 32 rows x 16 chunks of 16B
    const int row = c >> 4, cc = c & 15;
    ASYNC_CP_B128(gX + row * 64 + cc * 4, &Xs[row * 68 + cc * 4]);
  }
  for (int c = tid; c < 1024; c += 256) {  // 64x64 floats, linear
    ASYNC_CP_B128(W + c * 4, &Ws[c * 4]);
  }
  WAIT_ASYNC0();
  __syncthreads();

  // ---- WMMA from LDS ----
  const int lane = tid & 31;
  const int wave = tid >> 5;
  const int rt   = wave >> 2;        // row tile 0..1
  const int ct   = wave & 3;         // col tile 0..3
  const int mn   = lane & 15;        // A row / B,D column for this lane
  const int koff = (lane >> 4) << 1; // K pair select: 0 (lanes 0-15), 2 (16-31)

  const float* __restrict__ xr = &Xs[(rt * 16 + mn) * 68];
  const float* __restrict__ wc = &Ws[ct * 16 + mn];

  v8f acc = {};
  #pragma unroll
  for (int kk = 0; kk < 16; ++kk) {
    const int kb = kk * 4 + koff;
    v2f a; a.x = xr[kb];       a.y = xr[kb + 1];
    v2f b; b.x = wc[kb * 64];  b.y = wc[(kb + 1) * 64];
    acc = __builtin_amdgcn_wmma_f32_16x16x4_f32(
        /*neg_a=*/false, a, /*neg_b=*/false, b,
        /*c_mod=*/(short)0, acc, /*reuse_a=*/false, /*reuse_b=*/false);
  }
  // D layout: VGPR r -> (M = r + (lane>=16 ? 8 : 0), N = lane&15)
  const int rbase = row0 + rt * 16 + ((lane >> 4) << 3);
  #pragma unroll
  for (int r = 0; r < 8; ++r)
    Hout[(size_t)(rbase + r) * 64 + ct * 16 + mn] = acc[r];
}

// ---------- alpha_s / alpha_d per (node, head) ----------
__global__ void k_alpha(const float* __restrict__ H, const float* __restrict__ a_src,
                        const float* __restrict__ a_dst, float* __restrict__ as,
                        float* __restrict__ ad, int total /* BN*4 */) {
  int i = blockIdx.x * blockDim.x + threadIdx.x;
  if (i >= total) return;
  const int head = i & 3;
  const int bn   = i >> 2;
  const float* __restrict__ hp = H + (size_t)bn * 64 + head * 16;
  const float* __restrict__ sv = a_src + head * 16;
  const float* __restrict__ dv = a_dst + head * 16;
  float vs = 0.f, vd = 0.f;
  #pragma unroll
  for (int c = 0; c < 16; ++c) { float hv = hp[c]; vs += hv * sv[c]; vd += hv * dv[c]; }
  as[i] = vs;
  ad[i] = vd;
}

// ---------- edge pass 1: scatter-max of leaky_relu(alpha_s[src]+alpha_d[dst]) ----------
__global__ void k_edge_max(const int* __restrict__ ei, const float* __restrict__ as,
                           const float* __restrict__ ad, unsigned* __restrict__ emax,
                           int E, int Etot, int N, int total /* B*Etot */) {
  int idx = blockIdx.x * blockDim.x + threadIdx.x;
  if (idx >= total) return;
  const int b = idx / Etot;
  const int e = idx - b * Etot;
  int src, dst;
  if (e < E) { src = ei[e]; dst = ei[E + e]; } else { src = e - E; dst = src; }
  const size_t bs = ((size_t)b * N + src) * 4;
  const size_t bd = ((size_t)b * N + dst) * 4;
  #pragma unroll
  for (int h = 0; h < 4; ++h) {
    float v = lrelu(as[bs + h] + ad[bd + h]);
    atomicMax(&emax[bd + h], fflip(v));
  }
}

// ---------- edge pass 2: exp, denom += ee, acc[dst] += ee * h[src] ----------
__global__ void k_edge_accum(const int* __restrict__ ei, const float* __restrict__ as,
                             const float* __restrict__ ad, const unsigned* __restrict__ emax,
                             const float* __restrict__ Hm, float* __restrict__ denom,
                             float* __restrict__ acc, int E, int Etot, int N,
                             int total /* B*Etot*4 */) {
  int idx = blockIdx.x * blockDim.x + threadIdx.x;
  if (idx >= total) return;
  const int head = idx & 3;
  const int rest = idx >> 2;
  const int b = rest / Etot;
  const int e = rest - b * Etot;
  int src, dst;
  if (e < E) { src = ei[e]; dst = ei[E + e]; } else { src = e - E; dst = src; }
  const size_t bns = (size_t)b * N + src;
  const size_t bnd = (size_t)b * N + dst;
  float v  = lrelu(as[bns * 4 + head] + ad[bnd * 4 + head]);
  float m  = funflip(emax[bnd * 4 + head]);
  float ee = expf(v - m);
  atomicAdd(&denom[bnd * 4 + head], ee);
  const float* __restrict__ hs = Hm + bns * 64 + head * 16;
  float* __restrict__ ap = acc + bnd * 64 + head * 16;
  #pragma unroll
  for (int c = 0; c < 16; ++c) atomicAdd(&ap[c], ee * hs[c]);
}

// ---------- finalize: x_out = relu(acc/denom + bias) ----------
__global__ void k_finalize(const float* __restrict__ acc, const float* __restrict__ denom,
                           const float* __restrict__ bias, float* __restrict__ Xout,
                           int total /* BN*64 */) {
  int i = blockIdx.x * blockDim.x + threadIdx.x;
  if (i >= total) return;
  const int f = i & 63;
  const int bn = i >> 6;
  const int head = f >> 4;
  float d = denom[(size_t)bn * 4 + head];
  float v = acc[i] / (d + 1e-16f) + bias[f];
  Xout[i] = v > 0.f ? v : 0.f;
}

// ---------- readout: logits + per-batch max ----------
__global__ void k_logits(const float* __restrict__ emb, const float* __restrict__ w_out,
                         const float* __restrict__ b_out, float* __restrict__ logits,
                         unsigned* __restrict__ bmax, int N, int total /* B*N */) {
  int i = blockIdx.x * blockDim.x + threadIdx.x;
  if (i >= total) return;
  const float* __restrict__ e = emb + (size_t)i * 64;
  float v = b_out[0];
  #pragma unroll
  for (int f = 0; f < 64; ++f) v += e[f] * w_out[f];
  logits[i] = v;
  atomicMax(&bmax[i / N], fflip(v));
}

__global__ void k_sumexp(const float* __restrict__ logits, const unsigned* __restrict__ bmax,
                         float* __restrict__ bsum, int N, int total) {
  int i = blockIdx.x * blockDim.x + threadIdx.x;
  if (i >= total) return;
  float m = funflip(bmax[i / N]);
  atomicAdd(&bsum[i / N], expf(logits[i] - m));
}

__global__ void k_logsoftmax(const float* __restrict__ logits, const unsigned* __restrict__ bmax,
                             const float* __restrict__ bsum, float* __restrict__ out,
                             int N, int total) {
  int i = blockIdx.x * blockDim.x + threadIdx.x;
  if (i >= total) return;
  float m = funflip(bmax[i / N]);
  out[i] = logits[i] - m - logf(bsum[i / N]);
}

extern "C" void kernel_launch(void* const* d_in, const int* in_sizes, int n_in,
                              void* d_out, int out_size, void* d_ws, size_t ws_size,
                              hipStream_t stream) {
  const float* x    = (const float*)d_in[0];   // [B, N, 64]
  const int*   ei   = (const int*)d_in[1];     // [2, E]
  const float* W1   = (const float*)d_in[2];
  const float* a1s  = (const float*)d_in[3];
  const float* a1d  = (const float*)d_in[4];
  const float* b1   = (const float*)d_in[5];
  const float* W2   = (const float*)d_in[6];
  const float* a2s  = (const float*)d_in[7];
  const float* a2d  = (const float*)d_in[8];
  const float* b2   = (const float*)d_in[9];
  const float* wout = (const float*)d_in[10];  // [64,1]
  const float* bout = (const float*)d_in[11];  // [1]

  const int B    = 8;
  const int F    = 64;
  const int E    = in_sizes[1] / 2;
  const int N    = in_sizes[0] / (B * F);
  const int BN   = B * N;
  const int Etot = E + N;  // self loops appended

  // ---- workspace carve-up (256B aligned) ----
  char* ws = (char*)d_ws;
  size_t off = 0;
  auto carve = [&](size_t bytes) -> void* {
    void* p = ws + off;
    off = (off + bytes + 255) & ~(size_t)255;
    return p;
  };
  float*    h      = (float*)carve((size_t)BN * 64 * sizeof(float));   // gemm out / final emb
  float*    acc    = (float*)carve((size_t)BN * 64 * sizeof(float));   // scatter accumulator
  float*    xmid   = (float*)carve((size_t)BN * 64 * sizeof(float));   // layer-1 activations
  float*    as_    = (float*)carve((size_t)BN * 4  * sizeof(float));
  float*    ad_    = (float*)carve((size_t)BN * 4  * sizeof(float));
  unsigned* emax   = (unsigned*)carve((size_t)BN * 4 * sizeof(unsigned));
  float*    denom  = (float*)carve((size_t)BN * 4  * sizeof(float));
  float*    logits = (float*)carve((size_t)BN * sizeof(float));
  unsigned* bmax   = (unsigned*)carve(B * sizeof(unsigned));
  float*    bsum   = (float*)carve(B * sizeof(float));
  (void)ws_size; (void)n_in; (void)out_size;

  const int TB = 256;
  auto gat_layer = [&](const float* xin, const float* Wm, const float* av_s,
                       const float* av_d, const float* bias, float* xout) {
    k_gemm64<<<BN / 32, 256, 0, stream>>>(xin, Wm, h, BN);
    k_alpha<<<(BN * 4 + TB - 1) / TB, TB, 0, stream>>>(h, av_s, av_d, as_, ad_, BN * 4);
    (void)hipMemsetAsync(acc,   0, (size_t)BN * 64 * sizeof(float), stream);
    (void)hipMemsetAsync(denom, 0, (size_t)BN * 4  * sizeof(float), stream);
    (void)hipMemsetAsync(emax,  0, (size_t)BN * 4  * sizeof(unsigned), stream);  // key 0 == -inf
    const int t1 = B * Etot;
    k_edge_max<<<(t1 + TB - 1) / TB, TB, 0, stream>>>(ei, as_, ad_, emax, E, Etot, N, t1);
    const int t2 = B * Etot * 4;
    k_edge_accum<<<(t2 + TB - 1) / TB, TB, 0, stream>>>(ei, as_, ad_, emax, h, denom, acc,
                                                        E, Etot, N, t2);
    const int t3 = BN * 64;
    k_finalize<<<(t3 + TB - 1) / TB, TB, 0, stream>>>(acc, denom, bias, xout, t3);
  };

  // Layer 1: comm_data -> xmid ; Layer 2: xmid -> h (reuse; gemm output consumed before overwrite)
  gat_layer(x,    W1, a1s, a1d, b1, xmid);
  gat_layer(xmid, W2, a2s, a2d, b2, h);

  (void)hipMemsetAsync(bmax, 0, B * sizeof(unsigned), stream);
  (void)hipMemsetAsync(bsum, 0, B * sizeof(float), stream);
  k_logits<<<(BN + TB - 1) / TB, TB, 0, stream>>>(h, wout, bout, logits, bmax, N, BN);
  k_sumexp<<<(BN + TB - 1) / TB, TB, 0, stream>>>(logits, bmax, bsum, N, BN);
  k_logsoftmax<<<(BN + TB - 1) / TB, TB, 0, stream>>>(logits, bmax, bsum, (float*)d_out, N, BN);
}